// QuantumRefinementBlock_7576322310283
// MI455X (gfx1250) — compile-verified
//
#include <hip/hip_runtime.h>
#include <math.h>

typedef __attribute__((ext_vector_type(2))) float v2f;
typedef __attribute__((ext_vector_type(8))) float v8f;

#define NB   16      // batch
#define NC   256     // channels
#define HWN  16384   // 128*128

// ---------------------------------------------------------------------------
// Kernel 1: per-(b,c) plane mean.  One 256-thread block per 64KB plane.
// Reads x once from HBM (268MB) -> bandwidth bound.
// ---------------------------------------------------------------------------
__global__ void pool_kernel(const float* __restrict__ x, float* __restrict__ pooled) {
    __shared__ float red[256];
    const int p = blockIdx.x;                       // plane = b*256 + c
    const float4* xv = (const float4*)x + (size_t)p * (HWN / 4);
    float s = 0.f;
#pragma unroll
    for (int it = 0; it < 16; ++it) {
        float4 v = xv[threadIdx.x + it * 256];
        s += v.x + v.y + v.z + v.w;
    }
    red[threadIdx.x] = s;
    __syncthreads();
    for (int off = 128; off > 0; off >>= 1) {
        if (threadIdx.x < off) red[threadIdx.x] += red[threadIdx.x + off];
        __syncthreads();
    }
    if (threadIdx.x == 0) pooled[p] = red[0] * (1.0f / (float)HWN);
}

// ---------------------------------------------------------------------------
// Kernel 2: single wave32.  GEMM1 (WMMA f32 16x16x4, K=256) -> tanh ->
// 4-qubit statevector circuit (one batch per lane, registers) ->
// GEMM2 (WMMA f32 16x16x4, K=4) -> sigmoid gates[16][256].
// ---------------------------------------------------------------------------
__global__ void qgate_kernel(const float* __restrict__ pooled,
                             const float* __restrict__ enc_w,   // [4][256]
                             const float* __restrict__ enc_b,   // [4]
                             const float* __restrict__ qw,      // [2][4][3]
                             const float* __restrict__ proj_w,  // [256][4]
                             const float* __restrict__ proj_b,  // [256]
                             float* __restrict__ gates)         // [16][256]
{
    __shared__ float zbuf[16][16];
    __shared__ float refbuf[16][4];

    const int lane = threadIdx.x;     // 0..31
    const int m    = lane & 15;       // row (A) / col (B,C,D)
    const int hi   = lane >> 4;
    const int kOff = hi * 2;          // K pair owned by this half-wave

    // Branch-free B-operand setup: always load from the valid clamped row
    // (m & 3), then zero-select on the loaded VALUE (v_cndmask, not exec
    // branches).  Row base pointers are 8B-aligned for v2f (b64) loads.
    const bool bvalid = (m < 4);
    const float* aRow = pooled + m * 256 + kOff;
    const float* bRow = enc_w + (m & 3) * 256 + kOff;

    // ---- GEMM1: z_pre[16,16] = pooled[16,256] x enc_w^T (qubits padded to 16)
    v8f acc = {};
#pragma unroll 4
    for (int k0 = 0; k0 < 256; k0 += 4) {
        v2f a = *(const v2f*)(aRow + k0);
        v2f braw = *(const v2f*)(bRow + k0);
        v2f b;
        b.x = bvalid ? braw.x : 0.f;   // B[k][n] = enc_w[n][k], rows n>=4 are zero
        b.y = bvalid ? braw.y : 0.f;
        acc = __builtin_amdgcn_wmma_f32_16x16x4_f32(false, a, false, b,
                                                    (short)0, acc, false, false);
    }
#pragma unroll
    for (int j = 0; j < 8; ++j) zbuf[j + 8 * hi][m] = acc[j];   // D row=j+8*hi, col=m
    __syncthreads();

    // ---- Quantum circuit: lane t simulates batch t (16 complex amplitudes)
    if (lane < 16) {
        float z0[4];
#pragma unroll
        for (int q = 0; q < 4; ++q) z0[q] = tanhf(zbuf[lane][q] + enc_b[q]);

        float sr[16], si[16];
#pragma unroll
        for (int i = 0; i < 16; ++i) { sr[i] = 0.f; si[i] = 0.f; }
        sr[0] = 1.f;

        // AngleEmbedding: RY(z_w) on wire w (bit 3-w)
#pragma unroll
        for (int w = 0; w < 4; ++w) {
            float a  = 0.5f * z0[w];
            float cw = cosf(a), sw = sinf(a);
            int mask = 1 << (3 - w);
#pragma unroll
            for (int i = 0; i < 16; ++i) {
                if (i & mask) continue;
                int j = i | mask;
                float r0 = cw * sr[i] - sw * sr[j];
                float m0 = cw * si[i] - sw * si[j];
                float r1 = sw * sr[i] + cw * sr[j];
                float m1 = sw * si[i] + cw * si[j];
                sr[i] = r0; si[i] = m0; sr[j] = r1; si[j] = m1;
            }
        }

        // StronglyEntanglingLayers
#pragma unroll
        for (int l = 0; l < 2; ++l) {
#pragma unroll
            for (int w = 0; w < 4; ++w) {
                float phi = qw[(l * 4 + w) * 3 + 0];
                float th  = qw[(l * 4 + w) * 3 + 1];
                float om  = qw[(l * 4 + w) * 3 + 2];
                float ct = cosf(0.5f * th), st = sinf(0.5f * th);
                float h1 = 0.5f * (phi + om), h2 = 0.5f * (phi - om);
                float c1 = cosf(h1), s1 = sinf(h1);
                float c2 = cosf(h2), s2 = sinf(h2);
                // Rot = RZ(om) RY(th) RZ(phi)
                float u00r =  c1 * ct, u00i = -s1 * ct;
                float u01r = -c2 * st, u01i = -s2 * st;
                float u10r =  c2 * st, u10i = -s2 * st;
                float u11r =  c1 * ct, u11i =  s1 * ct;
                int mask = 1 << (3 - w);
#pragma unroll
                for (int i = 0; i < 16; ++i) {
                    if (i & mask) continue;
                    int j = i | mask;
                    float ar = sr[i], ai = si[i], br = sr[j], bi = si[j];
                    sr[i] = u00r * ar - u00i * ai + u01r * br - u01i * bi;
                    si[i] = u00r * ai + u00i * ar + u01r * bi + u01i * br;
                    sr[j] = u10r * ar - u10i * ai + u11r * br - u11i * bi;
                    si[j] = u10r * ai + u10i * ar + u11r * bi + u11i * br;
                }
            }
            int r = (l % 3) + 1;   // ring range
#pragma unroll
            for (int w = 0; w < 4; ++w) {
                int t  = (w + r) & 3;
                int mc = 1 << (3 - w), mt = 1 << (3 - t);
#pragma unroll
                for (int i = 0; i < 16; ++i) {
                    if ((i & mc) && !(i & mt)) {
                        int j = i ^ mt;
                        float tr = sr[i]; sr[i] = sr[j]; sr[j] = tr;
                        float ti = si[i]; si[i] = si[j]; si[j] = ti;
                    }
                }
            }
        }

        // PauliZ expvals
#pragma unroll
        for (int w = 0; w < 4; ++w) {
            int mask = 1 << (3 - w);
            float ev = 0.f;
#pragma unroll
            for (int i = 0; i < 16; ++i) {
                float p = sr[i] * sr[i] + si[i] * si[i];
                ev += (i & mask) ? -p : p;
            }
            refbuf[lane][w] = ev;
        }
    }
    __syncthreads();

    // ---- GEMM2: gates_pre[16,256] = refined[16,4] x proj_w^T ; sigmoid
    v2f a2;
    a2.x = refbuf[m][kOff];
    a2.y = refbuf[m][kOff + 1];
#pragma unroll
    for (int c0 = 0; c0 < 256; c0 += 16) {
        int ch = c0 + m;                       // this lane's output column
        v2f b2 = *(const v2f*)(proj_w + ch * 4 + kOff);   // B[k][n] = proj_w[ch][k]
        v8f cz = {};
        cz = __builtin_amdgcn_wmma_f32_16x16x4_f32(false, a2, false, b2,
                                                   (short)0, cz, false, false);
        float bias = proj_b[ch];
#pragma unroll
        for (int j = 0; j < 8; ++j) {
            float g = 1.0f / (1.0f + expf(-(cz[j] + bias)));
            gates[(j + 8 * hi) * 256 + ch] = g;
        }
    }
}

// ---------------------------------------------------------------------------
// Kernel 3: out = x * (1 + gate[b][c]).  x is re-read (268MB).
// ---------------------------------------------------------------------------
__global__ void apply_kernel(const float* __restrict__ x,
                             const float* __restrict__ gates,
                             float* __restrict__ out) {
    const int p = blockIdx.x;
    const float scale = 1.0f + gates[p];        // uniform per block -> scalar load
    const float4* xv = (const float4*)x   + (size_t)p * (HWN / 4);
    float4*       ov = (float4*)out       + (size_t)p * (HWN / 4);
#pragma unroll
    for (int it = 0; it < 16; ++it) {
        int idx = threadIdx.x + it * 256;
        float4 v = xv[idx];
        v.x *= scale; v.y *= scale; v.z *= scale; v.w *= scale;
        ov[idx] = v;
    }
}

extern "C" void kernel_launch(void* const* d_in, const int* in_sizes, int n_in,
                              void* d_out, int out_size, void* d_ws, size_t ws_size,
                              hipStream_t stream) {
    (void)in_sizes; (void)n_in; (void)out_size; (void)ws_size;
    const float* x      = (const float*)d_in[0];
    const float* enc_w  = (const float*)d_in[1];
    const float* enc_b  = (const float*)d_in[2];
    const float* qw     = (const float*)d_in[3];
    const float* proj_w = (const float*)d_in[4];
    const float* proj_b = (const float*)d_in[5];
    float* out    = (float*)d_out;
    float* pooled = (float*)d_ws;          // 4096 floats
    float* gates  = pooled + NB * NC;      // 4096 floats

    pool_kernel<<<NB * NC, 256, 0, stream>>>(x, pooled);
    qgate_kernel<<<1, 32, 0, stream>>>(pooled, enc_w, enc_b, qw, proj_w, proj_b, gates);
    apply_kernel<<<NB * NC, 256, 0, stream>>>(x, gates, out);
}